// FeatureContrast_62586263437916
// MI455X (gfx1250) — compile-verified
//
#include <hip/hip_runtime.h>

// FeatureContrast on MI455X (gfx1250).
// Memory-bound: 226 MB HBM traffic -> ~9.7us floor @ 23.3 TB/s.
// Strategy: stream channel rows; 3-wide row sums done as WMMA f32 16x16x4
// (banded constant A x exp-tile B) so the matrix pipe does the stencil adds
// while VALU/trans does v_exp_f32; rolling 3 D-tiles form the 3x3 denom.
// Output stores are non-temporal (never re-read).

typedef float v2f __attribute__((ext_vector_type(2)));
typedef float v8f __attribute__((ext_vector_type(8)));

#define N_GRID 16   // channel grid is 16x16
#define N_OUT  14   // valid interior 14x14
#define HW     4096 // 64*64 pixels per (b, channel) plane
#define C_IN   256
#define C_OUT  196
#define PIX_PER_WAVE 16
#define WAVES_PER_BLOCK 8

__global__ __launch_bounds__(256) void fc_wmma_kernel(
    const float* __restrict__ x, const float* __restrict__ weight,
    float* __restrict__ out)
{
  const int lane = threadIdx.x & 31;
  const int wave = threadIdx.x >> 5;
  const int t  = lane & 15;   // pixel within wave tile == N index
  const int hi = lane >> 4;   // lane half selects K / M block

  // Each wave owns 16 consecutive pixels; 4096 pixels per batch image,
  // 128 pixels per block -> never straddles a batch boundary.
  const int P0 = blockIdx.x * (WAVES_PER_BLOCK * PIX_PER_WAVE) + wave * PIX_PER_WAVE;
  const int b  = P0 >> 12;
  const int s0 = P0 & 4095;

  const float w = weight[0];

  const float* __restrict__ xb = x   + (size_t)b * (C_IN  * HW) + s0 + t;
  float*       __restrict__ ob = out + (size_t)b * (C_OUT * HW) + s0 + t;

  // Constant banded A: A[m,u] = 1 iff m<14 and u in {m, m+1, m+2}.
  // 16x4-f32 A layout: M = lane%16; chunk p: VGPR 2p holds K=4p+2*hi,
  // VGPR 2p+1 holds K=4p+1+2*hi.
  const int m = t;
  float a[8];
#pragma unroll
  for (int p = 0; p < 4; ++p) {
#pragma unroll
    for (int q = 0; q < 2; ++q) {
      int u = 4 * p + 2 * hi + q;
      a[2 * p + q] = (m < N_OUT && u >= m && u <= m + 2) ? 1.0f : 0.0f;
    }
  }

  v8f d[3]; // rolling ring of row3 tiles (channel rows gr-2, gr-1, gr)

#pragma unroll
  for (int gr = 0; gr < N_GRID; ++gr) {
    // Load E_gr = exp(w*x) in B-matrix (4x16 f32) layout, 4 K-chunks.
    // Each b32 load: lanes 0-15 contiguous 64B at channel c0,
    // lanes 16-31 contiguous 64B at channel c0+2 -> fully coalesced.
    float bb[8];
#pragma unroll
    for (int p = 0; p < 4; ++p) {
      int c0 = N_GRID * gr + 4 * p + 2 * hi;
      bb[2 * p]     = __expf(w * xb[c0 * HW]);
      bb[2 * p + 1] = __expf(w * xb[(c0 + 1) * HW]);
    }

    // row3[j,t] = sum_{u=j..j+2} e[u,t] via 4 chained 16x16x4 f32 WMMAs.
    v8f acc = {};
#pragma unroll
    for (int p = 0; p < 4; ++p) {
      v2f A; A.x = a[2 * p];  A.y = a[2 * p + 1];
      v2f B; B.x = bb[2 * p]; B.y = bb[2 * p + 1];
      acc = __builtin_amdgcn_wmma_f32_16x16x4_f32(
          false, A, false, B, (short)0, acc, false, false);
    }
    d[gr % 3] = acc;

    if (gr >= 2) {
      const int i = gr - 2; // output grid row; center channel row = gr-1
#pragma unroll
      for (int v = 0; v < 8; ++v) {
        int j = v + 8 * hi;                       // D layout: M = v + 8*hi
        float den = d[0][v] + d[1][v] + d[2][v];  // 3x3 window complete
        // Center value (channel row gr-1, col j+1) in D layout; this re-read
        // is temporally adjacent to the B-layout read -> L0/L2 hit, no extra HBM.
        float xc = xb[(N_GRID * (gr - 1) + j + 1) * HW];
        float res = xc * __expf(w * xc) / den;
        if (j < N_OUT) { // mask dead D rows 14,15 (lane-predicated store only)
          __builtin_nontemporal_store(res, ob + (i * N_OUT + j) * HW);
        }
      }
    }
  }
}

extern "C" void kernel_launch(void* const* d_in, const int* in_sizes, int n_in,
                              void* d_out, int out_size, void* d_ws, size_t ws_size,
                              hipStream_t stream) {
  const float* x      = (const float*)d_in[0]; // (32, 256, 64, 64) f32
  const float* weight = (const float*)d_in[1]; // (1,) f32
  float* out = (float*)d_out;                  // (32, 196, 64, 64) f32

  // 32*64*64 = 131072 pixels; 16 per wave, 8 waves per block -> 1024 blocks.
  dim3 grid(131072 / (WAVES_PER_BLOCK * PIX_PER_WAVE));
  dim3 block(256);
  fc_wmma_kernel<<<grid, block, 0, stream>>>(x, weight, out);
}